// DCRNN_38792144618012
// MI455X (gfx1250) — compile-verified
//
#include <hip/hip_runtime.h>
#include <math.h>

// Problem constants (from reference)
#define ROWS    5
#define COLS    5
#define NN      25          // ROWS*COLS
#define HID     4
#define NGATES  16          // 4*HID
#define T_STEPS 128
#define BS      8192
#define FFN     128         // NUM_NEURONS
#define HWID    200         // 2*NN*HID
#define LBS     64          // batch elements per block in recurrent kernel (2 waves)

typedef float v2f __attribute__((ext_vector_type(2)));
typedef float v8f __attribute__((ext_vector_type(8)));

#if __has_builtin(__builtin_amdgcn_global_load_async_to_lds_b32)
#define USE_ASYNC 1
typedef __attribute__((address_space(1))) int GI;   // global int* (builtin's src type)
typedef __attribute__((address_space(3))) int LI;   // LDS int*   (builtin's dst type)
#endif

__device__ __forceinline__ float fsig(float x) {
    // 1/(1+e^-x) : v_exp_f32 + v_rcp_f32
    return __builtin_amdgcn_rcpf(1.0f + __expf(-x));
}
__device__ __forceinline__ float ftanh(float x) {
    // tanh(x) = 1 - 2/(e^{2x}+1); saturates correctly at +/-1
    return 1.0f - 2.0f * __builtin_amdgcn_rcpf(__expf(2.0f * x) + 1.0f);
}

// ---------------------------------------------------------------------------
// Kernel 1: transpose x (bs, N, T) -> xT (T, N, bs) so the recurrent kernel
// reads lane-coalesced slices xT[t][n][b..b+31].
// ---------------------------------------------------------------------------
__global__ __launch_bounds__(256) void transpose_x_kernel(
    const float* __restrict__ x, float* __restrict__ xT)
{
    __shared__ float tile[32][33];
    const int n  = blockIdx.z;
    const int b0 = blockIdx.x * 32;
    const int t0 = blockIdx.y * 32;
    const int tx = threadIdx.x;      // 0..31
    const int ty = threadIdx.y;      // 0..7
#pragma unroll
    for (int j = 0; j < 32; j += 8) {
        tile[ty + j][tx] =
            x[((size_t)(b0 + ty + j) * NN + n) * T_STEPS + (t0 + tx)];
    }
    __syncthreads();
#pragma unroll
    for (int j = 0; j < 32; j += 8) {
        xT[((size_t)(t0 + ty + j) * NN + n) * BS + (b0 + tx)] =
            tile[tx][ty + j];
    }
}

// ---------------------------------------------------------------------------
// Kernel 2: bidirectional grid-LSTM recurrence. One thread = one batch
// element; h[25][4], c[25][4] live in registers; node loop fully unrolled so
// all neighbor indices are static. blockIdx.y selects direction.
// Next timestep's x slice is prefetched into LDS with CDNA5 async loads.
// ---------------------------------------------------------------------------
__global__ __launch_bounds__(LBS) void grid_lstm_kernel(
    const float* __restrict__ xT,
    const float* __restrict__ h0f, const float* __restrict__ c0f,
    const float* __restrict__ h0b, const float* __restrict__ c0b,
    const float* __restrict__ Wx_f, const float* __restrict__ Wh_f,
    const float* __restrict__ Wn_f, const float* __restrict__ b_f,
    const float* __restrict__ Wx_b, const float* __restrict__ Wh_b,
    const float* __restrict__ Wn_b, const float* __restrict__ b_b,
    float* __restrict__ H)
{
    const int  lt  = threadIdx.x;
    const int  b   = blockIdx.x * LBS + lt;
    const bool fwd = (blockIdx.y == 0);

    const float* h0 = fwd ? h0f : h0b;
    const float* c0 = fwd ? c0f : c0b;
    const float* Wx = fwd ? Wx_f : Wx_b;
    const float* Wh = fwd ? Wh_f : Wh_b;
    const float* Wn = fwd ? Wn_f : Wn_b;
    const float* Bb = fwd ? b_f  : b_b;

    // Hoist weights (uniform across the wave -> scalar-friendly)
    float wx[NGATES], wb[NGATES], wh[HID][NGATES], wn[HID][NGATES];
#pragma unroll
    for (int j = 0; j < NGATES; ++j) { wx[j] = Wx[j]; wb[j] = Bb[j]; }
#pragma unroll
    for (int k = 0; k < HID; ++k) {
#pragma unroll
        for (int j = 0; j < NGATES; ++j) {
            wh[k][j] = Wh[k * NGATES + j];
            wn[k][j] = Wn[k * NGATES + j];
        }
    }

    // State in registers: h0*/c0* are (N, bs, HID) -> float4 per node
    float h[NN][HID], c[NN][HID];
#pragma unroll
    for (int n = 0; n < NN; ++n) {
        float4 hv = *(const float4*)(h0 + ((size_t)n * BS + b) * HID);
        float4 cv = *(const float4*)(c0 + ((size_t)n * BS + b) * HID);
        h[n][0] = hv.x; h[n][1] = hv.y; h[n][2] = hv.z; h[n][3] = hv.w;
        c[n][0] = cv.x; c[n][1] = cv.y; c[n][2] = cv.z; c[n][3] = cv.w;
    }

#ifdef USE_ASYNC
    __shared__ float xbuf[2][NN * LBS];   // double-buffered x slice (12.8 KB)
    auto issue_x = [&](int tq, int s) {
        const float* src = xT + (size_t)tq * NN * BS + b;
        float* dst = &xbuf[s][lt];
#pragma unroll
        for (int n = 0; n < NN; ++n) {
            __builtin_amdgcn_global_load_async_to_lds_b32(
                (GI*)(src + (size_t)n * BS), (LI*)(dst + n * LBS), 0, 0);
        }
    };
    issue_x(fwd ? 0 : (T_STEPS - 1), 0);
#endif

#pragma unroll 1
    for (int t = 0; t < T_STEPS; ++t) {
        const int tt = fwd ? t : (T_STEPS - 1 - t);
#ifdef USE_ASYNC
        const int cur = t & 1;
        if (t + 1 < T_STEPS) {
            issue_x(fwd ? (t + 1) : (T_STEPS - 2 - t), cur ^ 1);
            // 25 newest outstanding belong to the next buffer; loads complete
            // in order, so <=25 guarantees the current buffer is resident.
            asm volatile("s_wait_asynccnt 25" ::: "memory");
        } else {
            asm volatile("s_wait_asynccnt 0" ::: "memory");
        }
#endif
#pragma unroll
        for (int i = 0; i < NN; ++i) {
#ifdef USE_ASYNC
            float xv = xbuf[cur][i * LBS + lt];
#else
            float xv = xT[((size_t)tt * NN + i) * BS + b];
#endif
            const float hs0 = h[i][0], hs1 = h[i][1], hs2 = h[i][2], hs3 = h[i][3];
            // Reference semantics incl. row-wrap for left/right (no col check)
            float nb_up = (i >= COLS)      ? h[i - COLS][3] : 0.0f;
            float nb_dn = (i < NN - COLS)  ? h[i + COLS][3] : 0.0f;
            float nb_lf = (i >= 1)         ? h[i - 1][3]    : 0.0f;
            float nb_rt = (i < NN - 1)     ? h[i + 1][3]    : 0.0f;
            float nb2 = fwd ? nb_lf : nb_rt;    // f:[up,dn,lf,rt] b:[up,dn,rt,lf]
            float nb3 = fwd ? nb_rt : nb_lf;

            float g[NGATES];
#pragma unroll
            for (int j = 0; j < NGATES; ++j) {
                float acc = fmaf(xv, wx[j], wb[j]);
                acc = fmaf(hs0,  wh[0][j], acc);
                acc = fmaf(hs1,  wh[1][j], acc);
                acc = fmaf(hs2,  wh[2][j], acc);
                acc = fmaf(hs3,  wh[3][j], acc);
                acc = fmaf(nb_up, wn[0][j], acc);
                acc = fmaf(nb_dn, wn[1][j], acc);
                acc = fmaf(nb2,   wn[2][j], acc);
                acc = fmaf(nb3,   wn[3][j], acc);
                g[j] = acc;
            }
#pragma unroll
            for (int k = 0; k < HID; ++k) {
                // gates split: ig=g[0:4], fg=g[4:8], og=g[8:12], cc=g[12:16]
                float nc = fsig(g[HID + k]) * c[i][k]
                         + fsig(g[k]) * ftanh(g[3 * HID + k]);
                float nh = fsig(g[2 * HID + k]) * ftanh(nc);
                c[i][k] = nc;
                h[i][k] = nh;
            }
        }
    }

    // H layout: (bs, 25, 8) with [:,:,0:4]=fwd, [:,:,4:8]=bwd, flattened
    const int off = fwd ? 0 : HID;
#pragma unroll
    for (int n = 0; n < NN; ++n) {
        float4 hv = make_float4(h[n][0], h[n][1], h[n][2], h[n][3]);
        *(float4*)(H + (size_t)b * HWID + n * (2 * HID) + off) = hv;
    }
}

// ---------------------------------------------------------------------------
// Kernel 3: ff = sigmoid(H @ W_ff + b_ff) via V_WMMA_F32_16X16X4_F32.
// One wave computes one 16x16 output tile; K=200 = 50 blocks of 4.
// f32 A(16x4): lanes0-15 hold K0,K1 in v0,v1 for row M=lane; lanes16-31 K2,K3.
// f32 B(4x16): v0 = rows {0 | 2} (lane<16 | lane>=16), v1 = rows {1 | 3}.
// f32 D(16x16): VGPR r -> M = r + 8*(lane>=16), N = lane&15.
// ---------------------------------------------------------------------------
__global__ __launch_bounds__(256) void ff_gemm_kernel(
    const float* __restrict__ Hmat, const float* __restrict__ Wff,
    const float* __restrict__ bff, float* __restrict__ ff)
{
    const int wave  = (blockIdx.x * blockDim.x + threadIdx.x) >> 5;
    const int lane  = threadIdx.x & 31;
    const int mt    = wave / (FFN / 16);
    const int nt    = wave % (FFN / 16);
    const int m0    = mt * 16;
    const int n0    = nt * 16;
    const int mn    = lane & 15;
    const int khalf = lane >> 4;          // 0: K{0,1}/rows{0,1}; 1: K{2,3}/rows{2,3}

    const float* arow = Hmat + (size_t)(m0 + mn) * HWID + khalf * 2;
    const float* bcol = Wff + (size_t)(khalf * 2) * FFN + (n0 + mn);

    v8f acc = {};
#pragma unroll 5
    for (int kb = 0; kb < HWID / 4; ++kb) {
        v2f a = *(const v2f*)(arow + kb * 4);       // H[m][kb*4+khalf*2 +{0,1}]
        v2f bfrag;
        bfrag.x = bcol[(size_t)(kb * 4 + 0) * FFN]; // Wff[kb*4+khalf*2+0][n]
        bfrag.y = bcol[(size_t)(kb * 4 + 1) * FFN]; // Wff[kb*4+khalf*2+1][n]
        acc = __builtin_amdgcn_wmma_f32_16x16x4_f32(
            false, a, false, bfrag, (short)0, acc, false, false);
    }

    const int   coln = n0 + mn;
    const float bv   = bff[coln];
#pragma unroll
    for (int r = 0; r < 8; ++r) {
        int row = m0 + r + khalf * 8;
        float v = acc[r] + bv;
        ff[(size_t)row * FFN + coln] = fsig(v);
    }
}

// ---------------------------------------------------------------------------
// Kernel 4: out = softmax(ff @ W_out + b_out), 2 classes. Thread per row.
// ---------------------------------------------------------------------------
__global__ __launch_bounds__(256) void head_kernel(
    const float* __restrict__ ff, const float* __restrict__ Wout,
    const float* __restrict__ bout, float* __restrict__ out)
{
    const int bidx = blockIdx.x * blockDim.x + threadIdx.x;
    const float* row = ff + (size_t)bidx * FFN;
    float l0 = bout[0], l1 = bout[1];
#pragma unroll 4
    for (int j = 0; j < FFN; j += 4) {
        float4 v = *(const float4*)(row + j);
        l0 = fmaf(v.x, Wout[(j + 0) * 2 + 0], l0);
        l1 = fmaf(v.x, Wout[(j + 0) * 2 + 1], l1);
        l0 = fmaf(v.y, Wout[(j + 1) * 2 + 0], l0);
        l1 = fmaf(v.y, Wout[(j + 1) * 2 + 1], l1);
        l0 = fmaf(v.z, Wout[(j + 2) * 2 + 0], l0);
        l1 = fmaf(v.z, Wout[(j + 2) * 2 + 1], l1);
        l0 = fmaf(v.w, Wout[(j + 3) * 2 + 0], l0);
        l1 = fmaf(v.w, Wout[(j + 3) * 2 + 1], l1);
    }
    float m  = fmaxf(l0, l1);
    float e0 = __expf(l0 - m);
    float e1 = __expf(l1 - m);
    float inv = __builtin_amdgcn_rcpf(e0 + e1);
    out[(size_t)bidx * 2 + 0] = e0 * inv;
    out[(size_t)bidx * 2 + 1] = e1 * inv;
}

// ---------------------------------------------------------------------------
extern "C" void kernel_launch(void* const* d_in, const int* in_sizes, int n_in,
                              void* d_out, int out_size, void* d_ws, size_t ws_size,
                              hipStream_t stream)
{
    const float* x     = (const float*)d_in[0];
    const float* h0f   = (const float*)d_in[1];
    const float* c0f   = (const float*)d_in[2];
    const float* h0b   = (const float*)d_in[3];
    const float* c0b   = (const float*)d_in[4];
    const float* Wx_f  = (const float*)d_in[5];
    const float* Wh_f  = (const float*)d_in[6];
    const float* Wn_f  = (const float*)d_in[7];
    const float* b_f   = (const float*)d_in[8];
    const float* Wx_b  = (const float*)d_in[9];
    const float* Wh_b  = (const float*)d_in[10];
    const float* Wn_b  = (const float*)d_in[11];
    const float* b_b   = (const float*)d_in[12];
    const float* W_ff  = (const float*)d_in[13];
    const float* b_ff  = (const float*)d_in[14];
    const float* W_out = (const float*)d_in[15];
    const float* b_out = (const float*)d_in[16];
    float* out = (float*)d_out;

    // Workspace layout (floats): xT[T*N*BS] | H[BS*200] | ff[BS*128]
    float* xT = (float*)d_ws;
    float* H  = xT + (size_t)T_STEPS * NN * BS;
    float* ff = H  + (size_t)BS * HWID;

    transpose_x_kernel<<<dim3(BS / 32, T_STEPS / 32, NN), dim3(32, 8), 0, stream>>>(x, xT);

    grid_lstm_kernel<<<dim3(BS / LBS, 2), dim3(LBS), 0, stream>>>(
        xT, h0f, c0f, h0b, c0b,
        Wx_f, Wh_f, Wn_f, b_f,
        Wx_b, Wh_b, Wn_b, b_b, H);

    // (8192/16)*(128/16) = 4096 tiles, 1 wave each, 8 waves/block -> 512 blocks
    ff_gemm_kernel<<<dim3((BS / 16) * (FFN / 16) / 8), dim3(256), 0, stream>>>(
        H, W_ff, b_ff, ff);

    head_kernel<<<dim3(BS / 256), dim3(256), 0, stream>>>(ff, W_out, b_out, out);

    (void)in_sizes; (void)n_in; (void)out_size; (void)ws_size;
}